// SelfAttention_31138512896808
// MI455X (gfx1250) — compile-verified
//
#include <hip/hip_runtime.h>

// Problem constants (B,S,E,H = 2,2048,1024,16; head structure is irrelevant
// after the algebraic collapse -- output row depends only on (b, e)).
#define B_ 2
#define S_ 2048
#define E_ 1024
#define NCH 64                    // S-chunks for stage-1 column reduction
#define ROWS_PER_CH (S_ / NCH)    // 32

typedef float v2f __attribute__((ext_vector_type(2)));
typedef float v8f __attribute__((ext_vector_type(8)));

// ---------------------------------------------------------------------------
// Stage 1: partial column sums of `values` over S.
// grid (E_/256, NCH, B_), block 256. Fully coalesced: 256 consecutive e per block.
__global__ void colsum_partial_kernel(const float* __restrict__ values,
                                      float* __restrict__ partial) {
    const int e = blockIdx.x * 256 + threadIdx.x;
    const int c = blockIdx.y;
    const int b = blockIdx.z;
    const float* p = values + (size_t)b * S_ * E_ + (size_t)c * ROWS_PER_CH * E_ + e;
    float s = 0.f;
#pragma unroll
    for (int r = 0; r < ROWS_PER_CH; ++r) s += p[(size_t)r * E_];
    partial[((size_t)b * NCH + c) * E_ + e] = s;
}

// ---------------------------------------------------------------------------
// Stage 2: fold partials -> colsum[b, e].  grid (E_/256, B_), block 256.
__global__ void colsum_final_kernel(const float* __restrict__ partial,
                                    float* __restrict__ colsum) {
    const int e = blockIdx.x * 256 + threadIdx.x;
    const int b = blockIdx.y;
    const float* p = partial + (size_t)b * NCH * E_ + e;
    float s = 0.f;
#pragma unroll
    for (int c = 0; c < NCH; ++c) s += p[(size_t)c * E_];
    colsum[b * E_ + e] = s;
}

// ---------------------------------------------------------------------------
// Stages 3 & 4: out[b, n] = x[b, :] @ W^T[:, n] + bscale * bias[n], b in {0,1},
// done as a padded 16x1024x1024 GEMM via V_WMMA_F32_16X16X4_F32 (full f32).
// One wave (32 threads) per 16-wide output column tile; grid = E_/16 blocks.
//
// A fragment (16x4 f32):  VGPR0 = K+0 (lanes 0-15) / K+2 (lanes 16-31), VGPR1 = K+1 / K+3.
// B fragment (4x16 f32):  VGPR0 = row K+0 across lanes 0-15 / row K+2 across lanes 16-31.
// C/D (16x16 f32, 8 VGPRs): VGPR r holds M=r (lanes 0-15) and M=8+r (lanes 16-31).
__global__ void gemv2_wmma_kernel(const float* __restrict__ x,    // [2, E_]
                                  const float* __restrict__ W,    // [E_, E_] row-major
                                  const float* __restrict__ bias, // [E_]
                                  float bscale,
                                  float* __restrict__ out) {      // [2, E_]
    const int lane  = threadIdx.x;          // 0..31
    const int n0    = blockIdx.x * 16;      // output column tile
    const int m     = lane & 15;            // A row owned by this lane
    const int khalf = (lane >> 4) << 1;     // 0 or 2: K sub-slot per lane half
    const int n     = lane & 15;            // B column owned by this lane

    // Rows 2..15 of the padded A matrix are zero: clamp the pointer (stays
    // in-bounds) and zero via multiply -- no EXEC divergence at the WMMA.
    const float  avalid = (m < B_) ? 1.0f : 0.0f;
    const float* arow   = x + (size_t)(m < B_ ? m : (B_ - 1)) * E_;
    const float* wrow   = W + (size_t)(n0 + n) * E_;

    v8f c = {};
    for (int k = 0; k < E_; k += 4) {
        v2f a = *(const v2f*)(arow + k + khalf);   // 8B aligned: k%4==0, khalf even
        a.x *= avalid;
        a.y *= avalid;
        v2f b = *(const v2f*)(wrow + k + khalf);   // B[kk][n] = W[n0+n][k+kk]
        // (neg_a, A, neg_b, B, c_mod, C, reuse_a, reuse_b)
        c = __builtin_amdgcn_wmma_f32_16x16x4_f32(false, a, false, b,
                                                  (short)0, c, false, false);
    }

    // D rows M=0,1 (batches 0,1) live in c[0], c[1] of lanes 0-15.
    if (lane < 16) {
        const float bb = bias[n0 + lane] * bscale;
        out[0 * E_ + n0 + lane] = c[0] + bb;
        out[1 * E_ + n0 + lane] = c[1] + bb;
    }
}

// ---------------------------------------------------------------------------
// Stage 5: out[b, q, :] = row[b, :] for all q.  float4 stores, 16 MB total.
// grid = (B_*S_*E_/4)/256, block 256.
__global__ void broadcast_out_kernel(const float4* __restrict__ row,  // [2, E_/4]
                                     float4* __restrict__ out) {
    const size_t i     = (size_t)blockIdx.x * 256 + threadIdx.x;  // float4 index
    const size_t per_b = (size_t)S_ * (E_ / 4);
    const int b  = (int)(i / per_b);
    const int e4 = (int)(i & (E_ / 4 - 1));
    out[i] = row[b * (E_ / 4) + e4];
}

// ---------------------------------------------------------------------------
extern "C" void kernel_launch(void* const* d_in, const int* in_sizes, int n_in,
                              void* d_out, int out_size, void* d_ws, size_t ws_size,
                              hipStream_t stream) {
    (void)in_sizes; (void)n_in; (void)out_size; (void)ws_size;

    // setup_inputs order: values, keys, query, Wv, bv, Wk, bk, Wq, bq, Wo, bo.
    // keys/query/Wk/bk/Wq/bq are mathematically dead (softmax rows sum to 1).
    const float* values = (const float*)d_in[0];
    const float* Wv     = (const float*)d_in[3];
    const float* bv     = (const float*)d_in[4];
    const float* Wo     = (const float*)d_in[9];
    const float* bo     = (const float*)d_in[10];
    float* out = (float*)d_out;

    float* ws      = (float*)d_ws;
    float* partial = ws;                              // B_*NCH*E_ floats (512 KB)
    float* colsum  = partial + (size_t)B_ * NCH * E_; // B_*E_
    float* vsum    = colsum + B_ * E_;                // B_*E_
    float* rowv    = vsum + B_ * E_;                  // B_*E_

    colsum_partial_kernel<<<dim3(E_ / 256, NCH, B_), 256, 0, stream>>>(values, partial);
    colsum_final_kernel<<<dim3(E_ / 256, B_), 256, 0, stream>>>(partial, colsum);
    // vsum[b,:] = colsum[b,:] @ Wv^T + S*bv   (bias folded: Σ_s (x@Wv^T + bv) )
    gemv2_wmma_kernel<<<E_ / 16, 32, 0, stream>>>(colsum, Wv, bv, (float)S_, vsum);
    // rowv[b,:] = vsum[b,:] @ Wo^T + bo
    gemv2_wmma_kernel<<<E_ / 16, 32, 0, stream>>>(vsum, Wo, bo, 1.0f, rowv);
    broadcast_out_kernel<<<(B_ * S_ * E_ / 4) / 256, 256, 0, stream>>>(
        (const float4*)rowv, (float4*)out);
}